// PatchSampler_53317724013215
// MI455X (gfx1250) — compile-verified
//
#include <hip/hip_runtime.h>
#include <hip/hip_bf16.h>

// Problem constants: B=16, K=32, C=3, H=W=128
#define BK   512       // B*K
#define NCH  3
#define HW   16384     // 128*128 (one plane = 64 KB of f32 -> one TDM tile into LDS)

typedef unsigned int       u32;
typedef unsigned long long u64;
typedef __attribute__((ext_vector_type(4))) unsigned int u32x4;
typedef __attribute__((ext_vector_type(8))) int          i32x8;
typedef __attribute__((ext_vector_type(4))) int          i32x4;
typedef __attribute__((ext_vector_type(4))) float        f32x4;

// ---------------------------------------------------------------------------
// Tensor Data Mover: DMA one contiguous 128x128 f32 plane (64 KB) from global
// memory into LDS. Descriptor per CDNA5 ISA D# layout (CDNA5 ISA):
//   group0: count=1 | lds_addr | global_addr[56:0] | type=2
//   group1: data_size=4B, tensor_dim0/1=128, tile_dim0/1=128,
//           dim0_stride=128, dim1_stride=16384
// Issued by one wave; completion tracked with TENSORcnt.
// NOTE: taking (uintptr_t)&lds_tile[0] makes the LDS address escape (ptrtoint),
// which also prevents GlobalOpt from folding the never-visibly-stored LDS reads.
// ---------------------------------------------------------------------------
__device__ __forceinline__ void tdm_load_plane(const float* gsrc, float* lds_tile) {
  u64 ga = (u64)(uintptr_t)gsrc;
  u32 la = (u32)(uintptr_t)lds_tile;   // low 32 bits of generic ptr = LDS byte addr
  u32x4 g0;
  g0[0] = 1u;                                            // count=1 (valid user D#)
  g0[1] = la;                                            // lds_addr
  g0[2] = (u32)ga;                                       // global_addr[31:0]
  g0[3] = (u32)((ga >> 32) & 0x1FFFFFFu) | 0x80000000u;  // global_addr[56:32] | type=2
  i32x8 g1;
  g1[0] = 0x00020000;          // data_size=2 (4 bytes); no multicast/iterate/pad
  g1[1] = (int)(128u << 16);   // tensor_dim0 = 128   (bits 63:48)
  g1[2] = (int)(128u << 16);   // tensor_dim1 = 128   (bits 95:80)
  g1[3] = (int)(128u << 16);   // tile_dim0   = 128   (bits 127:112)
  g1[4] = 128;                 // tile_dim1   = 128   (bits 143:128)
  g1[5] = 128;                 // tensor_dim0_stride = 128
  g1[6] = (int)0x40000000;     // tensor_dim1_stride = 16384 (bits 223:208)
  g1[7] = 0;
  i32x4 gz = (i32x4)0;
#if __clang_major__ >= 23
  i32x8 gz8 = (i32x8)0;        // amdgpu-toolchain (clang-23): 6-arg form
  __builtin_amdgcn_tensor_load_to_lds(g0, g1, gz, gz, gz8, 0);
#else                          // ROCm 7.2 (clang-22): 5-arg form
  __builtin_amdgcn_tensor_load_to_lds(g0, g1, gz, gz, 0);
#endif
}

// Inner stage: bilinear rotation sample of the LDS-resident source plane at
// intermediate integer pixel (xx,yy). Zero-padded OOB taps, clamped gathers.
// `t` must be the __shared__ array itself so clang emits ds_load_b32.
__device__ __forceinline__ float bilin_rot(const float* t,
                                           float ca, float sa, int xx, int yy) {
  const float inv = 1.0f / 128.0f;
  float xs = (2.0f * (float)xx + 1.0f) * inv - 1.0f;
  float ys = (2.0f * (float)yy + 1.0f) * inv - 1.0f;
  float gx = ca * xs - sa * ys;
  float gy = sa * xs + ca * ys;
  float ix = ((gx + 1.0f) * 128.0f - 1.0f) * 0.5f;
  float iy = ((gy + 1.0f) * 128.0f - 1.0f) * 0.5f;
  float x0f = floorf(ix), y0f = floorf(iy);
  float wx1 = ix - x0f,   wy1 = iy - y0f;
  float wx0 = 1.0f - wx1, wy0 = 1.0f - wy1;
  int x0 = (int)x0f, y0 = (int)y0f;
  float acc = 0.0f;
#pragma unroll
  for (int dy = 0; dy < 2; ++dy) {
#pragma unroll
    for (int dx = 0; dx < 2; ++dx) {
      int xi = x0 + dx, yi = y0 + dy;
      bool in = ((u32)xi < 128u) & ((u32)yi < 128u);
      int xc = min(max(xi, 0), 127);
      int yc = min(max(yi, 0), 127);
      float v = t[(yc << 7) + xc];                 // ds_load_b32 from LDS tile
      float w = (dx ? wx1 : wx0) * (dy ? wy1 : wy0);
      acc += in ? (w * v) : 0.0f;
    }
  }
  return acc;
}

// Inner stage (labels): nearest rotation sample. rintf == jnp.round (RNE).
__device__ __forceinline__ float nearest_rot(const float* t,
                                             float ca, float sa, int xx, int yy) {
  const float inv = 1.0f / 128.0f;
  float xs = (2.0f * (float)xx + 1.0f) * inv - 1.0f;
  float ys = (2.0f * (float)yy + 1.0f) * inv - 1.0f;
  float gx = ca * xs - sa * ys;
  float gy = sa * xs + ca * ys;
  float ix = ((gx + 1.0f) * 128.0f - 1.0f) * 0.5f;
  float iy = ((gy + 1.0f) * 128.0f - 1.0f) * 0.5f;
  int xn = (int)rintf(ix), yn = (int)rintf(iy);
  bool in = ((u32)xn < 128u) & ((u32)yn < 128u);
  int xc = min(max(xn, 0), 127);
  int yc = min(max(yn, 0), 127);
  float v = t[(yc << 7) + xc];
  return in ? v : 0.0f;
}

// One workgroup per (b,k,channel) plane. Fused rotate->flip->scale, single pass.
// Each thread owns 4 consecutive-x pixels -> one global_store_b128 per quad;
// a wave covers exactly one 128-pixel row per iteration (fully coalesced).
__global__ void __launch_bounds__(256)
warp_patches_kernel(const float* __restrict__ patches,
                    const float* __restrict__ angles,
                    const int*   __restrict__ flip_h,
                    const int*   __restrict__ flip_v,
                    const float* __restrict__ scales,
                    float*       __restrict__ out_p) {
  __shared__ float tile[HW];                  // 64 KB: whole source plane

  const int plane = (int)blockIdx.x;          // 0..1535
  const int bk    = plane / NCH;
  const float* src = patches + (size_t)plane * HW;
  float*       dst = out_p   + (size_t)plane * HW;

  if (threadIdx.x < 32u) {                    // wave 0 issues the DMA
    tdm_load_plane(src, &tile[0]);
    __builtin_amdgcn_s_wait_tensorcnt(0);
  }
  __syncthreads();
  asm volatile("" ::: "memory");              // TDM wrote LDS behind the compiler

  const float ang = angles[bk];
  const float s   = scales[bk];
  const bool  fh  = flip_h[bk] != 0;
  const bool  fv  = flip_v[bk] != 0;
  float sa, ca;
  sincosf(ang, &sa, &ca);

  const float inv = 1.0f / 128.0f;
  for (int i = 0; i < 16; ++i) {
    const int base = (i << 10) + ((int)threadIdx.x << 2);   // 4 consecutive pixels
    const int y  = base >> 7;
    const int xb = base & 127;
    // y-side of the outer (scale) bilinear is shared by the quad
    float ys  = (2.0f * (float)y + 1.0f) * inv - 1.0f;
    float jy  = ((s * ys + 1.0f) * 128.0f - 1.0f) * 0.5f;
    float y0f = floorf(jy);
    float wy1 = jy - y0f, wy0 = 1.0f - wy1;
    int   y0  = (int)y0f;
    f32x4 r;
#pragma unroll
    for (int j = 0; j < 4; ++j) {
      int x = xb + j;
      float xs  = (2.0f * (float)x + 1.0f) * inv - 1.0f;
      float jx  = ((s * xs + 1.0f) * 128.0f - 1.0f) * 0.5f;
      float x0f = floorf(jx);
      float wx1 = jx - x0f, wx0 = 1.0f - wx1;
      int   x0  = (int)x0f;
      float acc = 0.0f;
#pragma unroll
      for (int dy = 0; dy < 2; ++dy) {
#pragma unroll
        for (int dx = 0; dx < 2; ++dx) {
          int xi = x0 + dx, yi = y0 + dy;
          bool in = ((u32)xi < 128u) & ((u32)yi < 128u);
          int xh = fh ? 127 - xi : xi;        // fold flip into lattice coords
          int yv = fv ? 127 - yi : yi;
          float w = (dx ? wx1 : wx0) * (dy ? wy1 : wy0);
          float v = bilin_rot(tile, ca, sa, xh, yv);
          acc += in ? (w * v) : 0.0f;
        }
      }
      r[j] = acc;
    }
    *(f32x4*)(dst + base) = r;                // global_store_b128
  }
}

// One workgroup per (b,k) label plane; nearest∘nearest = single LDS gather/pixel.
__global__ void __launch_bounds__(256)
warp_labels_kernel(const float* __restrict__ labels,
                   const float* __restrict__ angles,
                   const int*   __restrict__ flip_h,
                   const int*   __restrict__ flip_v,
                   const float* __restrict__ scales,
                   float*       __restrict__ out_l) {
  __shared__ float tile[HW];

  const int bk = (int)blockIdx.x;             // 0..511
  const float* src = labels + (size_t)bk * HW;
  float*       dst = out_l  + (size_t)bk * HW;

  if (threadIdx.x < 32u) {
    tdm_load_plane(src, &tile[0]);
    __builtin_amdgcn_s_wait_tensorcnt(0);
  }
  __syncthreads();
  asm volatile("" ::: "memory");

  const float ang = angles[bk];
  const float s   = scales[bk];
  const bool  fh  = flip_h[bk] != 0;
  const bool  fv  = flip_v[bk] != 0;
  float sa, ca;
  sincosf(ang, &sa, &ca);

  const float inv = 1.0f / 128.0f;
  for (int i = 0; i < 16; ++i) {
    const int base = (i << 10) + ((int)threadIdx.x << 2);
    const int y  = base >> 7;
    const int xb = base & 127;
    float ys = (2.0f * (float)y + 1.0f) * inv - 1.0f;
    float jy = ((s * ys + 1.0f) * 128.0f - 1.0f) * 0.5f;
    int   yi = (int)rintf(jy);
    bool  iny = ((u32)yi < 128u);
    int   yc = min(max(yi, 0), 127);
    int   yv = fv ? 127 - yc : yc;
    f32x4 r;
#pragma unroll
    for (int j = 0; j < 4; ++j) {
      int x = xb + j;
      float xs = (2.0f * (float)x + 1.0f) * inv - 1.0f;
      float jx = ((s * xs + 1.0f) * 128.0f - 1.0f) * 0.5f;
      int   xi = (int)rintf(jx);
      bool  in2 = iny & ((u32)xi < 128u);
      int   xc = min(max(xi, 0), 127);
      int   xh = fh ? 127 - xc : xc;
      float v = nearest_rot(tile, ca, sa, xh, yv);
      r[j] = in2 ? v : 0.0f;
    }
    *(f32x4*)(dst + base) = r;
  }
}

extern "C" void kernel_launch(void* const* d_in, const int* in_sizes, int n_in,
                              void* d_out, int out_size, void* d_ws, size_t ws_size,
                              hipStream_t stream) {
  const float* patches = (const float*)d_in[0];   // [16,32,3,128,128] f32
  const float* labels  = (const float*)d_in[1];   // [16,32,1,128,128] f32
  const float* angles  = (const float*)d_in[2];   // [16,32] f32
  const int*   flip_h  = (const int*)d_in[3];     // [16,32] bool->int
  const int*   flip_v  = (const int*)d_in[4];     // [16,32] bool->int
  const float* scales  = (const float*)d_in[5];   // [16,32] f32

  float* out_p = (float*)d_out;                          // 25,165,824 floats
  float* out_l = out_p + (size_t)BK * NCH * HW;          //  8,388,608 floats

  warp_patches_kernel<<<BK * NCH, 256, 0, stream>>>(patches, angles, flip_h, flip_v,
                                                    scales, out_p);
  warp_labels_kernel<<<BK, 256, 0, stream>>>(labels, angles, flip_h, flip_v,
                                             scales, out_l);
  (void)in_sizes; (void)n_in; (void)out_size; (void)d_ws; (void)ws_size;
}